// HungarianMatcher_24670292148838
// MI455X (gfx1250) — compile-verified
//
#include <hip/hip_runtime.h>

typedef float v2f __attribute__((ext_vector_type(2)));
typedef float v8f __attribute__((ext_vector_type(8)));

#define B_    64
#define N_    900
#define M_    100
#define K_    365
#define KPAD  372      // padded K row stride in LDS: mult of 4, bank-conflict-free
#define BIGF  1e30f

// ---------------------------------------------------------------------------
// Kernel 1: fused cost matrix, stored transposed costT[b][m][n]
//   cost = 2*(-prob) + 5*L1 + 2*(-giou)
// cost_class gather done as one-hot WMMA (exact in f32): probs(16xK) x 1hot(Kx16)
// ---------------------------------------------------------------------------
__global__ __launch_bounds__(128)
void hm_cost_kernel(const float* __restrict__ logits,
                    const float* __restrict__ pboxes,
                    const int*   __restrict__ tlabels,
                    const float* __restrict__ tboxes,
                    float* __restrict__ costT)
{
  __shared__ float sig[16 * KPAD];   // sigmoid(logits) tile, zero-padded
  __shared__ int   lab[112];
  __shared__ float tb[M_ * 4];
  __shared__ float pb[16 * 4];

  const int b   = blockIdx.y;
  const int n0  = blockIdx.x * 16;
  const int tid = threadIdx.x;

  // stage: coalesced streaming read of 16 logit rows, apply sigmoid into LDS
  const float* lrow = logits + (size_t)b * N_ * K_;
  for (int idx = tid; idx < 16 * KPAD; idx += blockDim.x) {
    int r = idx / KPAD, k = idx - r * KPAD;
    int n = n0 + r;
    float v = 0.f;
    if (k < K_ && n < N_) {
      float x = lrow[(size_t)n * K_ + k];
      v = 1.0f / (1.0f + __expf(-x));
    }
    sig[r * KPAD + k] = v;
  }
  for (int m = tid; m < 112; m += blockDim.x)
    lab[m] = (m < M_) ? tlabels[b * M_ + m] : -1;
  for (int idx = tid; idx < M_ * 4; idx += blockDim.x)
    tb[idx] = tboxes[(size_t)b * M_ * 4 + idx];
  for (int idx = tid; idx < 16 * 4; idx += blockDim.x) {
    int n = n0 + (idx >> 2);
    pb[idx] = (n < N_) ? pboxes[((size_t)b * N_ + n) * 4 + (idx & 3)] : 0.f;
  }
  __syncthreads();

  const int wave = tid >> 5;
  const int lane = tid & 31;
  const int half = lane >> 4;   // which 16-lane half
  const int l16  = lane & 15;

  for (int mt = wave; mt < 7; mt += 4) {         // 7 m-tiles over 4 waves
    const int m0    = mt * 16;
    const int myLab = lab[m0 + l16];
    v8f acc = {0.f,0.f,0.f,0.f,0.f,0.f,0.f,0.f};
    const float* arow = &sig[l16 * KPAD];        // A: row n = l16
    for (int kb = 0; kb < KPAD; kb += 4) {       // 93 k-steps of k=4
      int kh = kb + half * 2;                    // A lane layout: lanes16-31 hold K=2,3
      v2f a;  a.x  = arow[kh];  a.y = arow[kh + 1];
      v2f bb; bb.x = (myLab == kh)     ? 1.f : 0.f;   // one-hot B fragment
              bb.y = (myLab == kh + 1) ? 1.f : 0.f;
      acc = __builtin_amdgcn_wmma_f32_16x16x4_f32(
              /*neg_a=*/false, a, /*neg_b=*/false, bb,
              /*c_mod=*/(short)0, acc, /*reuse_a=*/false, /*reuse_b=*/false);
    }
    // acc[j] = prob[n0 + j + 8*half][m0 + l16]  (C layout: VGPRj -> M=j(+8), lane -> N)
    const int m = m0 + l16;
    if (m < M_) {
      float t_cx = tb[m*4+0], t_cy = tb[m*4+1], t_w = tb[m*4+2], t_h = tb[m*4+3];
      float tx0 = t_cx - 0.5f*t_w, ty0 = t_cy - 0.5f*t_h;
      float tx1 = t_cx + 0.5f*t_w, ty1 = t_cy + 0.5f*t_h;
      float areaT = t_w * t_h;
      float* out = costT + ((size_t)b * M_ + m) * N_;
      #pragma unroll
      for (int j = 0; j < 8; ++j) {
        int rloc = j + half * 8;
        int n = n0 + rloc;
        if (n < N_) {
          float p_cx = pb[rloc*4+0], p_cy = pb[rloc*4+1];
          float p_w  = pb[rloc*4+2], p_h  = pb[rloc*4+3];
          float px0 = p_cx - 0.5f*p_w, py0 = p_cy - 0.5f*p_h;
          float px1 = p_cx + 0.5f*p_w, py1 = p_cy + 0.5f*p_h;
          float l1 = fabsf(p_cx-t_cx)+fabsf(p_cy-t_cy)+fabsf(p_w-t_w)+fabsf(p_h-t_h);
          float ix0 = fmaxf(px0, tx0), iy0 = fmaxf(py0, ty0);
          float ix1 = fminf(px1, tx1), iy1 = fminf(py1, ty1);
          float inter = fmaxf(ix1-ix0, 0.f) * fmaxf(iy1-iy0, 0.f);
          float uni = p_w*p_h + areaT - inter;
          float iou = inter / (uni + 1e-6f);
          float ex0 = fminf(px0, tx0), ey0 = fminf(py0, ty0);
          float ex1 = fmaxf(px1, tx1), ey1 = fmaxf(py1, ty1);
          float areaE = fmaxf(ex1-ex0, 0.f) * fmaxf(ey1-ey0, 0.f);
          float giou = iou - (areaE - uni) / (areaE + 1e-6f);
          out[n] = -2.0f * acc[j] + 5.0f * l1 - 2.0f * giou;
        }
      }
    }
  }
}

// ---------------------------------------------------------------------------
// Kernel 2: Jonker-Volgenant LSA, one workgroup (256 thr = 8 waves) per batch.
// All state in LDS; column relaxation + argmin parallelized over threads.
// ---------------------------------------------------------------------------
__global__ __launch_bounds__(256)
void hm_lsa_kernel(const float* __restrict__ costAll,
                   int* __restrict__ outQ, int* __restrict__ outT)
{
  const int b = blockIdx.x;
  const int t = threadIdx.x;
  const int T = 256;
  const float* cost = costAll + (size_t)b * M_ * N_;

  __shared__ float u_[M_];
  __shared__ float v_[N_];
  __shared__ float shortest[N_];
  __shared__ int   path[N_];
  __shared__ int   row4col[N_];
  __shared__ int   col4row[M_];
  __shared__ unsigned char remaining[N_];
  __shared__ unsigned char SR[M_];
  __shared__ float redv[256];
  __shared__ int   redi[256];
  __shared__ int   s_i, s_sink;
  __shared__ float s_minVal;

  for (int r = t; r < M_; r += T) { u_[r] = 0.f; col4row[r] = -1; }
  for (int c = t; c < N_; c += T) { v_[c] = 0.f; row4col[c] = -1; }
  __syncthreads();

  for (int cur = 0; cur < M_; ++cur) {
    for (int c = t; c < N_; c += T) { remaining[c] = 1; shortest[c] = BIGF; path[c] = 0; }
    for (int r = t; r < M_; r += T) SR[r] = 0;
    if (t == 0) { s_i = cur; s_minVal = 0.f; s_sink = -1; }
    __syncthreads();

    while (s_sink < 0) {
      const int   i  = s_i;
      const float mv = s_minVal;
      const float ui = u_[i];
      if (t == 0) SR[i] = 1;
      const float* crow = cost + (size_t)i * N_;
      float bm = BIGF; int bj = N_;
      for (int c = t; c < N_; c += T) {
        if (remaining[c]) {
          float nv = mv + crow[c] - ui - v_[c];
          if (nv < shortest[c]) { shortest[c] = nv; path[c] = i; }
          float sv = shortest[c];
          if (sv < bm || (sv == bm && c < bj)) { bm = sv; bj = c; }
        }
      }
      redv[t] = bm; redi[t] = bj;
      __syncthreads();
      for (int s = T >> 1; s > 0; s >>= 1) {      // lexicographic (val, idx) argmin
        if (t < s) {
          float ov = redv[t + s]; int oi = redi[t + s];
          if (ov < redv[t] || (ov == redv[t] && oi < redi[t])) { redv[t] = ov; redi[t] = oi; }
        }
        __syncthreads();
      }
      if (t == 0) {
        int j = redi[0];
        s_minVal = redv[0];
        remaining[j] = 0;
        int r = row4col[j];
        if (r < 0) s_sink = j; else s_i = r;
      }
      __syncthreads();
    }

    // dual updates (scipy rectangular_lsap semantics, pre-augmentation col4row)
    const float minVal = s_minVal;
    if (t == 0) u_[cur] += minVal;
    for (int r = t; r < M_; r += T) {
      if (r != cur && SR[r]) {
        int cj = col4row[r];
        cj = cj < 0 ? 0 : (cj > N_ - 1 ? N_ - 1 : cj);
        u_[r] += minVal - shortest[cj];
      }
    }
    for (int c = t; c < N_; c += T) {
      if (!remaining[c]) v_[c] += shortest[c] - minVal;
    }
    __syncthreads();

    // augment alternating path (serial, tiny)
    if (t == 0) {
      int j = s_sink;
      for (;;) {
        int i = path[j];
        row4col[j] = i;
        int nj = col4row[i];
        col4row[i] = j;
        if (i == cur) break;
        j = nj;
      }
    }
    __syncthreads();
  }

  // argsort by assigned query index (values distinct -> rank = count smaller)
  if (t < M_) {
    int q = col4row[t];
    int rank = 0;
    for (int r2 = 0; r2 < M_; ++r2) rank += (col4row[r2] < q) ? 1 : 0;
    outQ[b * M_ + rank] = q;
    outT[b * M_ + rank] = t;
  }
}

// ---------------------------------------------------------------------------
extern "C" void kernel_launch(void* const* d_in, const int* in_sizes, int n_in,
                              void* d_out, int out_size, void* d_ws, size_t ws_size,
                              hipStream_t stream) {
  const float* logits  = (const float*)d_in[0];   // (B,N,K)
  const float* pboxes  = (const float*)d_in[1];   // (B,N,4)
  const int*   tlabels = (const int*)  d_in[2];   // (B,M)
  const float* tboxes  = (const float*)d_in[3];   // (B,M,4)

  float* costT = (float*)d_ws;                    // (B,M,N) = ~23 MB
  int* outQ = (int*)d_out;                        // (B,M) query_idx
  int* outT = outQ + B_ * M_;                     // (B,M) target_idx

  dim3 grid1((N_ + 15) / 16, B_);
  hm_cost_kernel<<<grid1, 128, 0, stream>>>(logits, pboxes, tlabels, tboxes, costT);
  hm_lsa_kernel<<<B_, 256, 0, stream>>>(costT, outQ, outT);
}